// DecompGen_78658031059107
// MI455X (gfx1250) — compile-verified
//
#include <hip/hip_runtime.h>
#include <math.h>

// ---------------- WMMA f32 16x16x4 helpers (wave32) ----------------
typedef __attribute__((ext_vector_type(2))) float v2f;
typedef __attribute__((ext_vector_type(8))) float v8f;

__device__ __forceinline__ v8f wmma4(v2f a, v2f b, v8f c) {
  // D = A(16x4) * B(4x16) + C(16x16), fp32
  return __builtin_amdgcn_wmma_f32_16x16x4_f32(false, a, false, b, (short)0, c,
                                               false, false);
}

// A layout (16x4): lane l<16 -> M=l, {v0,v1} = K={0,1}; lane l+16 -> K={2,3}
// B layout (4x16): lane l<16 -> N=l, {v0,v1} = K={0,1}; lane l+16 -> K={2,3}
// C/D layout: vgpr g -> M = g (+8 for lanes 16..31), N = lane&15

// ---------------- generic GEMM with compile-time strides ----------------
// C[MxN] = A[MxK] * B[KxN]
// A element (m,k) at A[m*SAM + k] (k contiguous)
// B element (k,n) at B[k*SBK + n*SBN]
// C element (m,n) at C[m*SCM + n]
template <int SAM, int SBK, int SBN, int SCM, int KTOT>
__global__ __launch_bounds__(32) void gemm_t(const float* __restrict__ A,
                                             const float* __restrict__ Bm,
                                             float* __restrict__ C) {
  const int lane = threadIdx.x & 31;
  const int hf = lane >> 4;
  const int l15 = lane & 15;
  const int nBase = blockIdx.x * 16;
  const int mBase = blockIdx.y * 16;
  const float* arow = A + (size_t)(mBase + l15) * SAM + 2 * hf;
  const float* bcol = Bm + (size_t)(nBase + l15) * SBN + (size_t)(2 * hf) * SBK;
  v8f acc = {};
#pragma unroll 4
  for (int kk = 0; kk < KTOT; kk += 4) {
    v2f a, b;
    a.x = arow[kk];
    a.y = arow[kk + 1];
    b.x = bcol[(size_t)kk * SBK];
    b.y = bcol[(size_t)(kk + 1) * SBK];
    acc = wmma4(a, b, acc);
  }
#pragma unroll
  for (int g = 0; g < 8; ++g)
    C[(size_t)(mBase + g + 8 * hf) * SCM + nBase + l15] = acc[g];
}

// ---------------- z -> BN(batch stats)+lrelu(0.01), assemble latent -------
// blocks 0..127: BN channel; blocks 128..255: embedding gather columns
__global__ void latent_kernel(const float* __restrict__ z,
                              const float* __restrict__ g,
                              const float* __restrict__ be,
                              const float* __restrict__ emb,
                              const int* __restrict__ label,
                              float* __restrict__ latent) {
  __shared__ float s1[256];
  __shared__ float s2[256];
  const int t = threadIdx.x;  // batch index, 256 threads
  const int blk = blockIdx.x;
  if (blk < 128) {
    const int ch = blk;
    const float v = z[t * 128 + ch];
    s1[t] = v;
    s2[t] = v * v;
    __syncthreads();
    for (int s = 128; s > 0; s >>= 1) {
      if (t < s) { s1[t] += s1[t + s]; s2[t] += s2[t + s]; }
      __syncthreads();
    }
    const float m = s1[0] * (1.0f / 256.0f);
    const float var = s2[0] * (1.0f / 256.0f) - m * m;
    const float scale = g[ch] * rsqrtf(var + 1e-5f);
    const float shift = be[ch] - m * scale;
    float val = v * scale + shift;
    latent[t * 256 + ch] = (val >= 0.0f) ? val : 0.01f * val;
  } else {
    const int ch = blk - 128;
    latent[t * 256 + 128 + ch] = emb[label[t] * 128 + ch];
  }
}

// ---------------- generic BN(batch stats)+lrelu in place ----------------
// one block per channel o; element e of channel o is at
//   X[(e/L)*sB + (e%L)*sL + o*sO],  e in [0,count)
__global__ void bn_lrelu(float* __restrict__ X, const float* __restrict__ gamma,
                         const float* __restrict__ beta, int count, int L,
                         int sB, int sL, int sO, float slope) {
  __shared__ float s1[256];
  __shared__ float s2[256];
  const int t = threadIdx.x;
  const int o = blockIdx.x;
  float sum = 0.0f, sq = 0.0f;
  for (int e = t; e < count; e += 256) {
    const size_t idx = (size_t)(e / L) * sB + (size_t)(e % L) * sL +
                       (size_t)o * sO;
    const float v = X[idx];
    sum += v;
    sq += v * v;
  }
  s1[t] = sum;
  s2[t] = sq;
  __syncthreads();
  for (int s = 128; s > 0; s >>= 1) {
    if (t < s) { s1[t] += s1[t + s]; s2[t] += s2[t + s]; }
    __syncthreads();
  }
  const float inv_n = 1.0f / (float)count;
  const float m = s1[0] * inv_n;
  const float var = s2[0] * inv_n - m * m;
  const float scale = gamma[o] * rsqrtf(var + 1e-5f);
  const float shift = beta[o] - m * scale;
  for (int e = t; e < count; e += 256) {
    const size_t idx = (size_t)(e / L) * sB + (size_t)(e % L) * sL +
                       (size_t)o * sO;
    const float v = X[idx] * scale + shift;
    X[idx] = (v >= 0.0f) ? v : slope * v;
  }
}

// ---------------- ConvT1d k=16: x(B,128,16) -> y(B,256,31) ----------------
// X layout (B, 2048) with col = i*16 + j ; W(128,256,16) torch flat
// Y layout [l][b][o] = l*65536 + b*256 + o
__global__ __launch_bounds__(32) void convt2(const float* __restrict__ X,
                                             const float* __restrict__ W,
                                             float* __restrict__ Y) {
  const int lane = threadIdx.x & 31;
  const int hf = lane >> 4;
  const int l15 = lane & 15;
  const int oBase = blockIdx.x * 16;
  const int bBase = blockIdx.y * 16;
  const int l = blockIdx.z;  // 0..30
  int t0 = l - 15; if (t0 < 0) t0 = 0;
  int t1 = l;      if (t1 > 15) t1 = 15;
  const float* xrow = X + (size_t)(bBase + l15) * 2048 + (size_t)(2 * hf) * 16;
  const float* wcol = W + (size_t)(oBase + l15) * 16 + (size_t)(2 * hf) * 4096;
  v8f acc = {};
  for (int t = t0; t <= t1; ++t) {
    const int j = l - t;
#pragma unroll 4
    for (int kk = 0; kk < 128; kk += 4) {
      v2f a, b;
      a.x = xrow[kk * 16 + j];
      a.y = xrow[kk * 16 + 16 + j];
      b.x = wcol[(size_t)kk * 1024 + t];
      b.y = wcol[(size_t)kk * 1024 + 1024 + t];
      acc = wmma4(a, b, acc);
    }
  }
  float* yb = Y + (size_t)l * 65536 + oBase + l15;
#pragma unroll
  for (int g = 0; g < 8; ++g)
    yb[(size_t)(bBase + g + 8 * hf) * 256] = acc[g];
}

// ---------------- ConvT1d k=2 + bias + tanh: y(B,256,31)->z(B,512,32) -----
// Y layout [j][b][o]; W3(256,512,2) torch flat; Z layout [b][m][p]
__global__ __launch_bounds__(32) void convt3(const float* __restrict__ Y,
                                             const float* __restrict__ W3,
                                             const float* __restrict__ b3,
                                             float* __restrict__ Z) {
  const int lane = threadIdx.x & 31;
  const int hf = lane >> 4;
  const int l15 = lane & 15;
  const int pBase = blockIdx.x * 16;
  const int bBase = blockIdx.y * 16;
  const int m = blockIdx.z;  // 0..31
  v8f acc = {};
  for (int t = 0; t < 2; ++t) {
    const int j = m - t;
    if (j < 0 || j > 30) continue;  // uniform across wave
    const float* yrow = Y + (size_t)j * 65536 + (size_t)(bBase + l15) * 256 +
                        2 * hf;
    const float* wcol = W3 + (size_t)(pBase + l15) * 2 + t +
                        (size_t)(2 * hf) * 1024;
#pragma unroll 4
    for (int kk = 0; kk < 256; kk += 4) {
      v2f a, b;
      a.x = yrow[kk];
      a.y = yrow[kk + 1];
      b.x = wcol[(size_t)kk * 1024];
      b.y = wcol[(size_t)kk * 1024 + 1024];
      acc = wmma4(a, b, acc);
    }
  }
  const float bias = b3[pBase + l15];
#pragma unroll
  for (int g = 0; g < 8; ++g) {
    const float v = tanhf(acc[g] + bias);
    Z[(size_t)(bBase + g + 8 * hf) * 16384 + m * 512 + pBase + l15] = v;
  }
}

// ---------------- final: out[b,ch,y,x] = sum_r coef*c*h*w ----------------
// C (B,1536) col=r*3+ch ; H,W layouts [b][pos][r] = b*16384+pos*512+r
__global__ __launch_bounds__(32) void final_einsum(
    const float* __restrict__ C, const float* __restrict__ H,
    const float* __restrict__ Wt, const float* __restrict__ coef,
    float* __restrict__ Out) {
  const int lane = threadIdx.x & 31;
  const int hf = lane >> 4;
  const int l15 = lane & 15;
  const int nBase = blockIdx.x * 16;  // x tile (0 or 16)
  const int mBase = blockIdx.y * 16;  // rows over 96 = ch*32+y
  const int b = blockIdx.z;
  const int row = mBase + l15;
  const int ch = row >> 5;  // /32
  const int y = row & 31;
  const float* cb = C + (size_t)b * 1536 + ch + (size_t)(2 * hf) * 3;
  const float* hb = H + (size_t)b * 16384 + y * 512 + 2 * hf;
  const float* wb = Wt + (size_t)b * 16384 + (nBase + l15) * 512 + 2 * hf;
  const float* cf = coef + 2 * hf;
  v8f acc = {};
#pragma unroll 4
  for (int kk = 0; kk < 512; kk += 4) {
    v2f a, bf;
    a.x = cb[kk * 3] * cf[kk] * hb[kk];
    a.y = cb[kk * 3 + 3] * cf[kk + 1] * hb[kk + 1];
    bf.x = wb[kk];
    bf.y = wb[kk + 1];
    acc = wmma4(a, bf, acc);
  }
  float* ob = Out + (size_t)b * 3072 + nBase + l15;
#pragma unroll
  for (int g = 0; g < 8; ++g)
    ob[(size_t)(mBase + g + 8 * hf) * 32] = acc[g];
}

// ------------------------------ launch ------------------------------
extern "C" void kernel_launch(void* const* d_in, const int* in_sizes, int n_in,
                              void* d_out, int out_size, void* d_ws,
                              size_t ws_size, hipStream_t stream) {
  (void)in_sizes; (void)n_in; (void)out_size; (void)ws_size;
  const float* noise = (const float*)d_in[0];
  const int* label = (const int*)d_in[1];
  const float* lin_w = (const float*)d_in[2];
  // lin_b (d_in[3]) cancels inside batch-stats BN
  const float* bn0_g = (const float*)d_in[4];
  const float* bn0_b = (const float*)d_in[5];
  const float* emb = (const float*)d_in[6];
  const float* c_w1 = (const float*)d_in[7];
  // c_b1 cancels
  const float* c_g1 = (const float*)d_in[9];
  const float* c_be1 = (const float*)d_in[10];
  const float* h_w1 = (const float*)d_in[11];
  const float* h_g1 = (const float*)d_in[13];
  const float* h_be1 = (const float*)d_in[14];
  const float* h_w2 = (const float*)d_in[15];
  const float* h_g2 = (const float*)d_in[17];
  const float* h_be2 = (const float*)d_in[18];
  const float* h_w3 = (const float*)d_in[19];
  const float* h_b3 = (const float*)d_in[20];
  const float* w_w1 = (const float*)d_in[21];
  const float* w_g1 = (const float*)d_in[23];
  const float* w_be1 = (const float*)d_in[24];
  const float* w_w2 = (const float*)d_in[25];
  const float* w_g2 = (const float*)d_in[27];
  const float* w_be2 = (const float*)d_in[28];
  const float* w_w3 = (const float*)d_in[29];
  const float* w_b3 = (const float*)d_in[30];
  const float* coef = (const float*)d_in[31];
  float* out = (float*)d_out;

  float* ws = (float*)d_ws;
  float* z = ws;                  //   32768
  float* latent = z + 32768;      //   65536
  float* c_act = latent + 65536;  //  393216  (B x 1536)
  float* tmp1 = c_act + 393216;   //  524288  (B x 2048)
  float* tmp2 = tmp1 + 524288;    // 2031616  (31 x 256 x 256)
  float* h3 = tmp2 + 2031616;     // 4194304  (B x 32 x 512)
  float* w3 = h3 + 4194304;       // 4194304
  // total ~45.8 MB

  const dim3 wave(32);

  // 1) z = noise @ lin_w^T  (M=256,N=128,K=100)
  gemm_t<100, 1, 100, 128, 100>
      <<<dim3(8, 16), wave, 0, stream>>>(noise, lin_w, z);
  // 2) BN0 + lrelu(0.01) + embedding gather -> latent (256x256)
  latent_kernel<<<256, 256, 0, stream>>>(z, bn0_g, bn0_b, emb, label, latent);
  // 3) c pre-activation: latent @ c_w1  (N=1536), then BN+lrelu(0.2)
  gemm_t<256, 1536, 1, 1536, 256>
      <<<dim3(96, 16), wave, 0, stream>>>(latent, c_w1, c_act);
  bn_lrelu<<<512, 256, 0, stream>>>(c_act, c_g1, c_be1, 768, 3, 1536, 1, 3,
                                    0.2f);

  // ---- h stack ----
  gemm_t<256, 2048, 1, 2048, 256>
      <<<dim3(128, 16), wave, 0, stream>>>(latent, h_w1, tmp1);
  bn_lrelu<<<128, 256, 0, stream>>>(tmp1, h_g1, h_be1, 4096, 16, 2048, 1, 16,
                                    0.2f);
  convt2<<<dim3(16, 16, 31), wave, 0, stream>>>(tmp1, h_w2, tmp2);
  bn_lrelu<<<256, 256, 0, stream>>>(tmp2, h_g2, h_be2, 7936, 256, 65536, 256,
                                    1, 0.2f);
  convt3<<<dim3(32, 16, 32), wave, 0, stream>>>(tmp2, h_w3, h_b3, h3);

  // ---- w stack (reuses tmp1/tmp2) ----
  gemm_t<256, 2048, 1, 2048, 256>
      <<<dim3(128, 16), wave, 0, stream>>>(latent, w_w1, tmp1);
  bn_lrelu<<<128, 256, 0, stream>>>(tmp1, w_g1, w_be1, 4096, 16, 2048, 1, 16,
                                    0.2f);
  convt2<<<dim3(16, 16, 31), wave, 0, stream>>>(tmp1, w_w2, tmp2);
  bn_lrelu<<<256, 256, 0, stream>>>(tmp2, w_g2, w_be2, 7936, 256, 65536, 256,
                                    1, 0.2f);
  convt3<<<dim3(32, 16, 32), wave, 0, stream>>>(tmp2, w_w3, w_b3, w3);

  // ---- fused rank-weighted outer-product contraction ----
  final_einsum<<<dim3(2, 6, 256), wave, 0, stream>>>(c_act, h3, w3, coef, out);
}